// CrossEntropy_19602230739911
// MI455X (gfx1250) — compile-verified
//
#include <hip/hip_runtime.h>

typedef float v2f __attribute__((ext_vector_type(2)));
typedef float v8f __attribute__((ext_vector_type(8)));

#define NITER 384   // chunk-pairs: 768 chunks of 16 ref pixels, 2-way n-split

// Low 32 bits of a generic pointer into LDS == LDS byte offset (AMDGPU flat
// lowering: generic = {aperture_hi, lds_offset}; HW truncates to addr[31:0]).
__device__ __forceinline__ unsigned lds_off32(const void* p) {
    return (unsigned)(unsigned long long)p;
}

// CDNA5 async DMA: copy data from global memory straight into LDS (ASYNCcnt).
__device__ __forceinline__ void async_copy_b128(unsigned lds, const float* g) {
    asm volatile("global_load_async_to_lds_b128 %0, %1, off"
                 :: "v"(lds), "v"(g) : "memory");
}
__device__ __forceinline__ void async_copy_b32(unsigned lds, const float* g) {
    asm volatile("global_load_async_to_lds_b32 %0, %1, off"
                 :: "v"(lds), "v"(g) : "memory");
}
__device__ __forceinline__ void wait_async_0() {
    asm volatile("s_wait_asynccnt 0x0" ::: "memory");
}

// B=2, NREF=3, F=256, H*W=4096, D=16, N=12288
// Flash-attention-style fused label-propagation cross entropy, exact fp32 via
// V_WMMA_F32_16X16X4_F32. Block = 4 waves: 2 t-subtiles x 2 n-halves.
__global__ __launch_bounds__(128, 1)
void label_prop_ce_kernel(const float* __restrict__ ref,
                          const float* __restrict__ tgt,
                          const float* __restrict__ lab,
                          const int*   __restrict__ tlabel,
                          float*       __restrict__ out)
{
    const int b    = blockIdx.x >> 7;      // 128 target tiles (32 px) per batch
    const int tile = blockIdx.x & 127;
    const int tid  = threadIdx.x;
    const int w    = tid >> 5;             // wave 0..3
    const int tw   = w & 1;                // t-subtile (16 columns)
    const int ns   = w >> 1;               // n-half: even / odd chunks
    const int ln   = tid & 31;
    const int c16  = ln & 15;
    const int h    = ln >> 4;

    extern __shared__ float a_lds[];       // 4 x 4096 floats: ref chunks [f][pix]
    __shared__ float lab_s[4][256];        // 4 x 1KB: label chunks, transposed [pix][d]
    __shared__ float sm2[4][16];
    __shared__ float sl2[4][16];
    __shared__ float sacc[4][16][16];

    const int t0 = tile * 32 + tw * 16;

    // ---- Q tile (16 target pixels x 256 feats) in registers, B-operand layout:
    // lane%16 = column t, VGPR j holds K = 2*h + j  ->  f = 4*kk + 2*h + j
    const float* tb = tgt + ((size_t)b << 20) + t0 + c16;      // b*256*4096
    v2f qreg[64];
    #pragma unroll
    for (int kk = 0; kk < 64; ++kk) {
        const float* qp = tb + (size_t)(4 * kk + 2 * h) * 4096;
        qreg[kk].x = qp[0];
        qreg[kk].y = qp[4096];
    }

    const float* refb = ref + ((size_t)(b * 3) << 20);
    const float* labb = lab + ((size_t)(b * 3) << 16);
    const unsigned lds0 = lds_off32(a_lds);
    const unsigned lab0 = lds_off32(&lab_s[0][0]);

    // Stage one chunk: 16KB ref tile (1024 x b128 units) + 1KB transposed label
    // tile (256 x b32 units, scattered) into LDS buffer bufi.
    auto stage = [&](int c, int bufi) {
        const int r  = c >> 8;              // ref image 0..2
        const int p0 = (c & 255) << 4;      // pixel offset
        const float* gb = refb + ((size_t)r << 20) + p0;
        const unsigned lb = lds0 + (unsigned)bufi * 16384u;
        #pragma unroll
        for (int s2 = 0; s2 < 8; ++s2) {
            const int uu = tid + s2 * 128;  // unit: f = uu>>2, pix4 = (uu&3)*4
            const float* g = gb + ((size_t)(uu >> 2) << 12) + ((uu & 3) << 2);
            async_copy_b128(lb + (unsigned)uu * 16u, g);
        }
        const float* gl = labb + ((size_t)r << 16) + p0;
        const unsigned ll = lab0 + (unsigned)bufi * 1024u;
        #pragma unroll
        for (int s2 = 0; s2 < 2; ++s2) {
            const int e   = tid + s2 * 128; // e = d*16 + pix
            const int d   = e >> 4;
            const int pix = e & 15;
            async_copy_b32(ll + (unsigned)((pix << 4) + d) * 4u,
                           gl + ((size_t)d << 12) + pix);   // store transposed
        }
    };

    stage(0, 0);
    stage(1, 1);

    float m = -3.0e38f;
    float l = 0.0f;
    v8f acc = {};   // pred accumulator, C layout: d = 8*h+v, t = c16

    for (int i = 0; i < NITER; ++i) {
        wait_async_0();        // buffers for this iteration are resident
        __syncthreads();       // everyone done reading buffers we overwrite next
        if (i + 1 < NITER) {   // prefetch next chunk pair while computing
            const int par = (i + 1) & 1;
            stage(2 * (i + 1),     par * 2);
            stage(2 * (i + 1) + 1, par * 2 + 1);
        }

        const int buf = (i & 1) * 2 + ns;
        const float* abase = a_lds + buf * 4096 + h * 32 + c16; // f=4kk+2h, pix=c16

        // ---- S tile (16 n x 16 t): 64 x wmma f32 16x16x4, two independent
        // accumulator chains so loads and WMMAs pipeline ----
        v8f s0 = {}, s1 = {};
        #pragma unroll
        for (int kb = 0; kb < 8; ++kb) {
            float ax[8], ay[8];
            #pragma unroll
            for (int u = 0; u < 8; ++u) {
                ax[u] = abase[(kb * 8 + u) * 64];
                ay[u] = abase[(kb * 8 + u) * 64 + 16];
            }
            #pragma unroll
            for (int u = 0; u < 8; u += 2) {
                v2f a0; a0.x = ax[u];     a0.y = ay[u];
                v2f a1; a1.x = ax[u + 1]; a1.y = ay[u + 1];
                s0 = __builtin_amdgcn_wmma_f32_16x16x4_f32(false, a0, false,
                        qreg[kb * 8 + u],     (short)0, s0, false, false);
                s1 = __builtin_amdgcn_wmma_f32_16x16x4_f32(false, a1, false,
                        qreg[kb * 8 + u + 1], (short)0, s1, false, false);
            }
        }
        const v8f s = s0 + s1;

        // ---- Online softmax update over the 16 n-rows of this chunk ----
        float mc = s[0];
        #pragma unroll
        for (int v = 1; v < 8; ++v) mc = fmaxf(mc, s[v]);
        mc = fmaxf(mc, __shfl_xor(mc, 16, 32));
        const float mnew = fmaxf(m, mc);
        const float corr = __expf(m - mnew);
        l *= corr;
        #pragma unroll
        for (int v = 0; v < 8; ++v) acc[v] *= corr;

        float p[8], q[8];
        float ps = 0.0f;
        #pragma unroll
        for (int v = 0; v < 8; ++v) { p[v] = __expf(s[v] - mnew); ps += p[v]; }
        ps += __shfl_xor(ps, 16, 32);
        l += ps;
        m = mnew;
        #pragma unroll
        for (int v = 0; v < 8; ++v) q[v] = __shfl_xor(p[v], 16, 32);
        // lane now sees full column: rows 8*h+v in p[], rows 8*(1-h)+v in q[]

        // ---- acc += lab_tile (16d x 16n) @ P (16n x 16t) in 4 wmma steps ----
        // A operand from transposed LDS tile: (pix = 4*st+2*h+{0,1}, d = c16)
        const float* lbase = &lab_s[buf][h * 32 + c16];
        #pragma unroll
        for (int st = 0; st < 4; ++st) {
            v2f pb;   // B operand rows k = 4*st + 2*h + {0,1}, col = c16
            if (st < 2) {
                pb.x = h ? q[4 * st + 2] : p[4 * st];
                pb.y = h ? q[4 * st + 3] : p[4 * st + 1];
            } else {
                pb.x = h ? p[4 * st - 6] : q[4 * st - 8];
                pb.y = h ? p[4 * st - 5] : q[4 * st - 7];
            }
            v2f la;
            la.x = lbase[st * 64];
            la.y = lbase[st * 64 + 16];
            acc = __builtin_amdgcn_wmma_f32_16x16x4_f32(false, la, false, pb,
                                                        (short)0, acc, false, false);
        }
    }

    // ---- Merge the two n-half states (waves w and w+2 share a t-subtile) ----
    if (ln < 16) { sm2[w][ln] = m; sl2[w][ln] = l; }
    #pragma unroll
    for (int v = 0; v < 8; ++v) sacc[w][8 * h + v][c16] = acc[v];
    __syncthreads();

    if (tid < 32) {
        const int t   = tid & 15;
        const int twi = tid >> 4;
        const int wa = twi, wb = twi + 2;
        const float Ma = sm2[wa][t], Mb = sm2[wb][t];
        const float M  = fmaxf(Ma, Mb);
        const float ea = __expf(Ma - M), eb = __expf(Mb - M);
        const float L  = sl2[wa][t] * ea + sl2[wb][t] * eb;
        const int dstar = tlabel[((size_t)b << 12) + tile * 32 + twi * 16 + t];
        const float P  = sacc[wa][dstar][t] * ea + sacc[wb][dstar][t] * eb;
        const float pred = P / L;
        atomicAdd(out, -__logf(pred + 1e-14f) * (1.0f / 8192.0f)); // mean over 8192
    }
}

extern "C" void kernel_launch(void* const* d_in, const int* in_sizes, int n_in,
                              void* d_out, int out_size, void* d_ws, size_t ws_size,
                              hipStream_t stream) {
    const float* ref = (const float*)d_in[0];   // (2,3,256,64,64)
    const float* tgt = (const float*)d_in[1];   // (2,256,64,64)
    const float* lab = (const float*)d_in[2];   // (2,3,16,64,64)
    const int*   tl  = (const int*)d_in[3];     // (2,64,64)
    float* out = (float*)d_out;

    hipMemsetAsync(out, 0, sizeof(float), stream);
    label_prop_ce_kernel<<<dim3(256), dim3(128), 4 * 4096 * sizeof(float), stream>>>(
        ref, tgt, lab, tl, out);
}